// Sampler_20581483282433
// MI455X (gfx1250) — compile-verified
//
#include <hip/hip_runtime.h>
#include <stdint.h>

#define BSZ    1024                       // rows (B)
#define VOC    50257                      // vocab (V)
#define NROUND 32
#define TPB    512                        // 16 wave32 per block
#define NWAVE  (TPB / 32)                 // 16
#define IPT    100                        // elements/thread; 400B => 16B-aligned segs
#define VPAD   (TPB * IPT)                // 51200 (>= VOC, padding zeroed)

__device__ __forceinline__ float wave_incl_scan(float v, int lane) {
    for (int d = 1; d < 32; d <<= 1) {
        float t = __shfl_up(v, d, 32);
        if (lane >= d) v += t;
    }
    return v;
}

__global__ __launch_bounds__(TPB)
void sampler_kernel(const float* __restrict__ logits,
                    const float* __restrict__ temperature,
                    const float* __restrict__ top_p,
                    const float* __restrict__ uniforms,   // [NROUND, BSZ]
                    const int*   __restrict__ top_k,
                    int* __restrict__ out)
{
    extern __shared__ char smem_raw[];
    float* probs = (float*)smem_raw;            // VPAD floats, 16B aligned
    float* scanA = probs + VPAD;                // TPB floats: per-thread inclusive sums
    float* waveS = scanA + TPB;                 // NWAVE floats: wave sums
    int*   waveC = (int*)(waveS + NWAVE);       // NWAVE ints:   wave counts
    int*   shI   = waveC + NWAVE;               // [0]=t*, [1]=idx
    float* shF   = (float*)(shI + 2);           // [0]=pv

    const int tid  = threadIdx.x;
    const int lane = tid & 31;
    const int wid  = tid >> 5;
    const int row  = blockIdx.x;

    // ---------------- Stage 1: async copy logits row -> LDS ----------------
    {
        const unsigned lds0    = (unsigned)(uintptr_t)probs;
        const unsigned row_off = (unsigned)row * (unsigned)(VOC * 4u);
        if ((row & 3) == 0) {
            // 16B-aligned row: bulk b128 async copies + 1-dword tail
            const int NQ = VOC / 4;                       // 12564
            for (int c = tid; c < NQ; c += TPB) {
                unsigned goff = row_off + (unsigned)c * 16u;
                unsigned loff = lds0 + (unsigned)c * 16u;
                asm volatile("global_load_async_to_lds_b128 %0, %1, %2"
                             :: "v"(loff), "v"(goff), "s"(logits) : "memory");
            }
            if (tid == 0) {
                unsigned goff = row_off + (unsigned)NQ * 16u;
                unsigned loff = lds0 + (unsigned)NQ * 16u;
                asm volatile("global_load_async_to_lds_b32 %0, %1, %2"
                             :: "v"(loff), "v"(goff), "s"(logits) : "memory");
            }
        } else {
            // misaligned row stride (VOC*4 % 16 == 4): dword async copies
            for (int e = tid; e < VOC; e += TPB) {
                unsigned goff = row_off + (unsigned)e * 4u;
                unsigned loff = lds0 + (unsigned)e * 4u;
                asm volatile("global_load_async_to_lds_b32 %0, %1, %2"
                             :: "v"(loff), "v"(goff), "s"(logits) : "memory");
            }
        }
        for (int j = VOC + tid; j < VPAD; j += TPB) probs[j] = 0.0f;
        asm volatile("s_wait_asynccnt 0x0" ::: "memory");
    }
    __syncthreads();

    // ---------------- Stage 2: softmax (max, exp-sum, normalize) ------------
    const float invT = 1.0f / temperature[0];
    float m = -INFINITY;
    for (int i = tid; i < VOC; i += TPB) m = fmaxf(m, probs[i]);
    for (int d = 16; d; d >>= 1) m = fmaxf(m, __shfl_xor(m, d, 32));
    if (lane == 0) waveS[wid] = m;
    __syncthreads();
    m = waveS[0];
    for (int w = 1; w < NWAVE; ++w) m = fmaxf(m, waveS[w]);
    __syncthreads();

    float z = 0.0f;
    for (int i = tid; i < VOC; i += TPB) {
        float e = __expf((probs[i] - m) * invT);
        probs[i] = e;
        z += e;
    }
    for (int d = 16; d; d >>= 1) z += __shfl_xor(z, d, 32);
    if (lane == 0) waveS[wid] = z;
    __syncthreads();
    z = 0.0f;
    for (int w = 0; w < NWAVE; ++w) z += waveS[w];
    const float invZ = 1.0f / z;
    __syncthreads();

    // Normalize + round-1 per-thread segment sums (pivot = 0), b128 traffic.
    float sseg = 0.0f;
    {
        float4* P4 = (float4*)(probs + tid * IPT);
        for (int q = 0; q < IPT / 4; ++q) {
            float4 v = P4[q];
            v.x *= invZ; sseg += v.x;
            v.y *= invZ; sseg += v.y;
            v.z *= invZ; sseg += v.z;
            v.w *= invZ; sseg += v.w;
            P4[q] = v;
        }
    }
    __syncthreads();

    // Combined block scan (of s) + block count reduce: 2 barriers.
    // Leaves scanA[t] = inclusive prefix (needed by the sequential walk).
    float incl, excl, total; int cnt_gt;
    auto combined = [&](float s, int cnt) {
        float ws  = wave_incl_scan(s, lane);
        float wex = __shfl_up(ws, 1, 32); if (lane == 0) wex = 0.0f;
        for (int d = 16; d; d >>= 1) cnt += __shfl_xor(cnt, d, 32);
        if (lane == 31) waveS[wid] = ws;
        if (lane == 0)  waveC[wid] = cnt;
        __syncthreads();
        float base = 0.0f, tot = 0.0f; int ct = 0;
        for (int w = 0; w < NWAVE; ++w) {          // broadcast LDS reads
            float sw = waveS[w];
            if (w < wid) base += sw;
            tot += sw;
            ct  += waveC[w];
        }
        incl = ws + base;
        excl = wex + base;
        total = tot;
        cnt_gt = ct;
        scanA[tid] = incl;
        __syncthreads();
    };

    combined(sseg, 0);

    const float TP = top_p[0];
    const int   TK = top_k[0];
    float pivot = 0.0f;
    int   token = 0;

    // ---------------- Stage 3: rejection-sampling rounds --------------------
    for (int r = 0; r < NROUND; ++r) {
        const float u      = uniforms[(size_t)r * BSZ + row];
        const float target = u * total;

        if (tid == 0) shI[0] = TPB;
        __syncthreads();
        if (incl >= target && excl < target) atomicMin(&shI[0], tid);
        __syncthreads();

        if (tid == 0) {
            int idx;
            int ts = shI[0];
            if (!(target > 0.0f)) {
                idx = 0;                                  // target <= 0 (or NaN)
            } else if (ts >= TPB) {
                idx = VOC - 1;                            // FP edge: no crossing
            } else {
                float run = (ts > 0) ? scanA[ts - 1] : 0.0f;
                int j = ts * IPT, jend = j + IPT;
                idx = jend - 1;
                for (; j < jend; ++j) {
                    float p = probs[j];
                    if (p > pivot) run += p;
                    if (run >= target) { idx = j; break; }
                }
                if (idx > VOC - 1) idx = VOC - 1;
            }
            shI[1] = idx;
            shF[0] = probs[idx];
        }
        __syncthreads();

        const int   idx = shI[1];
        const float pv  = shF[0];
        token = idx;

        // Fused sweep: acceptance stats for this round == truncated segment
        // sums for the next round (threshold pv). b128, conflict-free.
        int cnt = 0; float s2 = 0.0f;
        {
            const float4* P4 = (const float4*)(probs + tid * IPT);
            for (int q = 0; q < IPT / 4; ++q) {
                float4 v = P4[q];
                if (v.x > pv) { cnt++; s2 += v.x; }
                if (v.y > pv) { cnt++; s2 += v.y; }
                if (v.z > pv) { cnt++; s2 += v.z; }
                if (v.w > pv) { cnt++; s2 += v.w; }
            }
        }
        combined(s2, cnt);                 // -> incl/excl/total(=sum_gt)/cnt_gt

        const bool accept = (cnt_gt < TK) && (total < TP);
        if (accept) break;                 // uniform branch (shared-derived)
        pivot = pv;
    }

    if (tid == 0) out[row] = token;
}

extern "C" void kernel_launch(void* const* d_in, const int* in_sizes, int n_in,
                              void* d_out, int out_size, void* d_ws, size_t ws_size,
                              hipStream_t stream) {
    (void)in_sizes; (void)n_in; (void)out_size; (void)d_ws; (void)ws_size;
    const float* logits = (const float*)d_in[0];
    const float* temp   = (const float*)d_in[1];
    const float* topp   = (const float*)d_in[2];
    const float* uni    = (const float*)d_in[3];
    const int*   topk   = (const int*)d_in[4];
    int* out = (int*)d_out;

    size_t shmem = (size_t)(VPAD + TPB + NWAVE) * sizeof(float)  // probs + scanA + waveS
                 + (size_t)NWAVE * sizeof(int)                   // waveC
                 + 4 * sizeof(int);                              // shI/shF scalars
    sampler_kernel<<<BSZ, TPB, shmem, stream>>>(logits, temp, topp, uni, topk, out);
}